// ProtoPNetDist_53102975648266
// MI455X (gfx1250) — compile-verified
//
#include <hip/hip_runtime.h>
#include <stdint.h>

typedef float v2f __attribute__((ext_vector_type(2)));
typedef float v8f __attribute__((ext_vector_type(8)));
typedef int   v4i __attribute__((ext_vector_type(4)));

#define B_  64
#define S_  1024
#define E_  1600
#define P_  100
#define Kp  4
#define PK  400
#define C_  2

#define NT  25          // 400 / 16 column tiles
#define EK  16          // E-chunk staged in LDS
#define BS_STRIDE 20    // padded LDS row stride (floats): bank-conflict-free AND 16B-aligned rows
#define BS_BUF (PK * BS_STRIDE)   // 8000 floats = 32 KB per buffer

// workspace layout (floats/uints):
//   [0, 400)            : p2buf   (float)  sum_e protolayer^2 per (p,k)
//   [400, 400 + 25600)  : minbuf  (uint bits of non-negative float) per (b,p,k)

// ---- CDNA5 async global->LDS copy (ASYNCcnt-tracked, no VGPR round trip) ----
__device__ __forceinline__ void async_ld_b128(const float* g, float* l) {
#if __has_builtin(__builtin_amdgcn_global_load_async_to_lds_b128)
  __attribute__((address_space(1))) v4i* gp =
      (__attribute__((address_space(1))) v4i*)const_cast<float*>(g);
  __attribute__((address_space(3))) v4i* lp =
      (__attribute__((address_space(3))) v4i*)l;
  __builtin_amdgcn_global_load_async_to_lds_b128(gp, lp, 0, 0);
#else
  unsigned lds = (unsigned)(uintptr_t)l;  // low 32 bits of shared-aperture addr = LDS byte address
  asm volatile("global_load_async_to_lds_b128 %0, %1, off"
               :: "v"(lds), "v"((unsigned long long)(uintptr_t)g)
               : "memory");
#endif
}

__device__ __forceinline__ void wait_async0() {
#if __has_builtin(__builtin_amdgcn_s_wait_asynccnt)
  __builtin_amdgcn_s_wait_asynccnt(0);
#else
  asm volatile("s_wait_asynccnt 0x0" ::: "memory");
#endif
}

__global__ __launch_bounds__(256) void p2_kernel(const float* __restrict__ proto,
                                                 float* __restrict__ p2buf) {
  int pk = blockIdx.x;
  const float* row = proto + (size_t)pk * E_;
  float s = 0.f;
  for (int e = threadIdx.x; e < E_; e += 256) { float v = row[e]; s += v * v; }
  for (int off = 16; off > 0; off >>= 1) s += __shfl_down(s, off, 32);
  __shared__ float red[8];
  int wave = threadIdx.x >> 5, lane = threadIdx.x & 31;
  if (lane == 0) red[wave] = s;
  __syncthreads();
  if (threadIdx.x == 0) {
    float t = 0.f;
    for (int i = 0; i < 8; ++i) t += red[i];
    p2buf[pk] = t;
  }
}

__global__ void init_min_kernel(unsigned int* __restrict__ minbuf) {
  int i = blockIdx.x * 256 + threadIdx.x;
  if (i < B_ * PK) minbuf[i] = 0x7F800000u;  // +inf
}

// One workgroup = one (b, 128-row S chunk). 8 waves; each wave owns a 16-row
// S tile and all 25 column tiles (acc[25] of v8f lives in registers).
// B (protolayer) E-chunks stream through a double-buffered LDS tile filled by
// async global->LDS B128 DMA that overlaps the WMMA compute of the prior chunk.
__global__ __launch_bounds__(256) void dist_kernel(const float* __restrict__ emb,
                                                   const float* __restrict__ proto,
                                                   const float* __restrict__ p2buf,
                                                   unsigned int* __restrict__ minbuf,
                                                   float* __restrict__ out) {
  __shared__ __align__(16) float Bs[2 * BS_BUF];  // 64 KB: two 400x16 chunks (padded rows)
  __shared__ float p2s[PK];

  const int b    = blockIdx.x;
  const int sc   = blockIdx.y;
  const int tid  = threadIdx.x;
  const int wave = tid >> 5;
  const int lane = tid & 31;
  const int half = lane >> 4;   // 0: K=e,e+1   1: K=e+2,e+3 (A-frag layout)
  const int lr   = lane & 15;
  const int s0   = sc * 128 + wave * 16;

  for (int i = tid; i < PK; i += 256) p2s[i] = p2buf[i];

  // A fragment base: row s0+lr of embedding[b], offset by half*2 within each K=4 step
  const float* Arow = emb + ((size_t)b * S_ + (size_t)(s0 + lr)) * E_ + half * 2;

  v8f acc[NT];
  const v8f vz = {0.f, 0.f, 0.f, 0.f, 0.f, 0.f, 0.f, 0.f};
#pragma unroll
  for (int n = 0; n < NT; ++n) acc[n] = vz;
  float a2 = 0.f;  // per-lane partial of sum_e x^2 (lanes l and l+16 cover a full row)

  // Stage one 400x16 B-chunk into LDS buffer `nbuf` via async B128 DMA.
  // 1600 B128 transfers (4 per pk row of 16 floats); rows padded to 20 floats (80 B, 16B-aligned).
  auto stage = [&](int ec, int nbuf) {
    const float* src = proto + ec * EK;
    float* dst = &Bs[nbuf * BS_BUF];
    for (int i = tid; i < PK * 4; i += 256) {
      int row = i >> 2, q = i & 3;
      async_ld_b128(src + (size_t)row * E_ + q * 4, dst + row * BS_STRIDE + q * 4);
    }
  };

  stage(0, 0);
  wait_async0();
  __syncthreads();  // chunk 0 visible everywhere (also covers p2s)

  for (int ec = 0; ec < E_ / EK; ++ec) {
    const int cur = ec & 1;
    if (ec + 1 < E_ / EK) stage(ec + 1, cur ^ 1);  // DMA next chunk during compute
    const float* bs = &Bs[cur * BS_BUF];
#pragma unroll
    for (int st = 0; st < 4; ++st) {
      v2f a = *(const v2f*)(Arow + ec * EK + st * 4);
      a2 += a.x * a.x + a.y * a.y;
#pragma unroll
      for (int n = 0; n < NT; ++n) {
        // B frag (4x16, K x N): lane holds W[pk = n*16+lr][e + 2*half + {0,1}]
        v2f bf = *(const v2f*)(&bs[(n * 16 + lr) * BS_STRIDE + st * 4 + half * 2]);
        acc[n] = __builtin_amdgcn_wmma_f32_16x16x4_f32(
            /*neg_a=*/false, a, /*neg_b=*/false, bf,
            /*c_mod=*/(short)0, acc[n], /*reuse_a=*/false, /*reuse_b=*/false);
      }
    }
    wait_async0();    // this wave's DMA for chunk ec+1 landed
    __syncthreads();  // everyone's reads of chunk ec done + everyone's DMA visible
  }

  // ||x||^2 per row: combine the two half-lanes, then gather the 8 rows this
  // lane needs in the C layout (m = r + 8*half).
  float x2row = a2 + __shfl_xor(a2, 16, 32);
  float x2need[8];
#pragma unroll
  for (int r = 0; r < 8; ++r) x2need[r] = __shfl(x2row, r + half * 8, 32);

  float* dout = out + 6400;  // distances region: [B,P,S,K]
#pragma unroll
  for (int n = 0; n < NT; ++n) {
    const int pk = n * 16 + lr;
    const int p = pk >> 2, k = pk & 3;
    const float p2v = p2s[pk];
    float mind = 3.4e38f;
    const size_t base = (((size_t)b * P_ + p) * S_) * Kp + k;
#pragma unroll
    for (int r = 0; r < 8; ++r) {
      float d2 = x2need[r] + p2v - 2.0f * acc[n][r];
      float d = sqrtf(fmaxf(d2, 0.f));
      int s = s0 + r + half * 8;
      dout[base + (size_t)s * Kp] = d;
      mind = fminf(mind, d);
    }
    mind = fminf(mind, __shfl_xor(mind, 16, 32));
    if (half == 0)  // distances >= 0, so uint compare == float compare
      atomicMin(&minbuf[b * PK + pk], __float_as_uint(mind));
  }
}

__global__ __launch_bounds__(128) void final_kernel(const unsigned int* __restrict__ minbuf,
                                                    const float* __restrict__ fc,
                                                    float* __restrict__ out) {
  int b = blockIdx.x, t = threadIdx.x;
  __shared__ float pd[P_];
  if (t < P_) {
    float s = 0.f;
#pragma unroll
    for (int k = 0; k < Kp; ++k) s += __uint_as_float(minbuf[b * PK + t * Kp + k]);
    pd[t] = s;
    out[b * P_ + t] = s;  // prototype_distances [B,P]
  }
  __syncthreads();
  if (t < C_) {
    float s = 0.f;
    for (int p = 0; p < P_; ++p) s += pd[p] * fc[t * P_ + p];
    out[6400 + (size_t)B_ * P_ * S_ * Kp + (size_t)b * C_ + t] = s;  // class_out [B,C]
  }
}

extern "C" void kernel_launch(void* const* d_in, const int* in_sizes, int n_in,
                              void* d_out, int out_size, void* d_ws, size_t ws_size,
                              hipStream_t stream) {
  (void)in_sizes; (void)n_in; (void)out_size; (void)ws_size;
  const float* emb   = (const float*)d_in[0];  // [64,1024,1600]
  const float* proto = (const float*)d_in[1];  // [100,4,1600]
  const float* fc    = (const float*)d_in[2];  // [2,100]
  float* out = (float*)d_out;
  float* p2buf = (float*)d_ws;
  unsigned int* minbuf = (unsigned int*)d_ws + PK;

  p2_kernel<<<PK, 256, 0, stream>>>(proto, p2buf);
  init_min_kernel<<<(B_ * PK + 255) / 256, 256, 0, stream>>>(minbuf);
  dist_kernel<<<dim3(B_, S_ / 128), 256, 0, stream>>>(emb, proto, p2buf, minbuf, out);
  final_kernel<<<B_, 128, 0, stream>>>(minbuf, fc, out);
}